// HybridMemory_5600637354001
// MI455X (gfx1250) — compile-verified
//
#include <hip/hip_runtime.h>
#include <hip/hip_bf16.h>

// Problem constants (from the reference)
#define LROWS 8192   // memory bank rows
#define DDIM  2048   // feature dim
#define NROWS 4096   // batch rows
#define MOM   0.2f
#define TEMP_INV 20.0f   // 1/0.05

typedef __attribute__((ext_vector_type(2))) float v2f;
typedef __attribute__((ext_vector_type(8))) float v8f;

#define ASTRIDE 2052  // 2048 + 4 pad floats -> 16 lanes land on distinct bank pairs

// ---------------------------------------------------------------------------
// Kernel 0: clear the loss accumulator (d_out is poisoned by the harness)
// ---------------------------------------------------------------------------
__global__ void zero_loss_kernel(float* loss) {
    if (blockIdx.x == 0 && threadIdx.x == 0) loss[0] = 0.0f;
}

// ---------------------------------------------------------------------------
// Kernel 1: fused GEMM (f32 WMMA 16x16x4) + online log-softmax cross-entropy.
// Block = 4 waves, owns a 32-row strip of `features` staged in 256.5 KB LDS.
// Each wave: 2x2 register blocking (rows {0-15},{16-31} x 2 col tiles), and
// sweeps a disjoint quarter of the L=8192 bank columns. Per-row (max,sum,tgt)
// merged across the 4 waves through LDS at the end.
// ---------------------------------------------------------------------------
__global__ __launch_bounds__(128, 1)
void sim_loss_kernel(const float* __restrict__ feats,
                     const int*   __restrict__ pids,
                     const float* __restrict__ mem,
                     float*       __restrict__ loss_out) {
    extern __shared__ float aLds[];               // 32 * ASTRIDE floats (dynamic)
    __shared__ float wmax_s[4][32];
    __shared__ float wsum_s[4][32];
    __shared__ float wtgt_s[4][32];
    __shared__ int   prow[32];

    const int tid  = threadIdx.x;
    const int wav  = tid >> 5;                    // 0..3
    const int lane = tid & 31;                    // wave32 lane
    const int half = lane >> 4;                   // 0: rows r, 1: rows r+8
    const int l16  = lane & 15;
    const int row0 = blockIdx.x * 32;

    // --- cooperative stage of the 32-row A strip into padded LDS ---
    {
        const float4* src = (const float4*)(feats + (size_t)row0 * DDIM);
        for (int t = tid; t < 32 * (DDIM / 4); t += 128) {
            const int r = t >> 9;                 // DDIM/4 = 512 float4 per row
            const int c = t & 511;
            *(float4*)(aLds + r * ASTRIDE + c * 4) = src[(size_t)r * 512 + c];
        }
    }
    if (tid < 32) prow[tid] = pids[row0 + tid];
    __syncthreads();

    // rows owned by this lane: strip s, accumulator row r -> m = s*16 + r + 8*half
    int myTgt[2][8];
    #pragma unroll
    for (int s = 0; s < 2; ++s)
        #pragma unroll
        for (int r = 0; r < 8; ++r) myTgt[s][r] = prow[s * 16 + r + 8 * half];

    float run_max[2][8], run_sum[2][8], tgt[2][8];
    #pragma unroll
    for (int s = 0; s < 2; ++s)
        #pragma unroll
        for (int r = 0; r < 8; ++r) {
            run_max[s][r] = -3.0e38f; run_sum[s][r] = 0.0f; tgt[s][r] = 0.0f;
        }

    // --- sweep this wave's quarter of the bank columns, 32 columns at a time ---
    const int colBase = wav * (LROWS / 4);
    for (int col0 = colBase; col0 < colBase + LROWS / 4; col0 += 32) {
        v8f c00 = {0.f,0.f,0.f,0.f,0.f,0.f,0.f,0.f};
        v8f c01 = c00, c10 = c00, c11 = c00;
        const float* a0 = aLds + l16 * ASTRIDE + 2 * half;          // rows 0-15
        const float* a1 = aLds + (16 + l16) * ASTRIDE + 2 * half;   // rows 16-31
        const float* b0 = mem + (size_t)(col0 + l16) * DDIM + 2 * half;
        const float* b1 = mem + (size_t)(col0 + 16 + l16) * DDIM + 2 * half;
        #pragma unroll 4
        for (int k = 0; k < DDIM; k += 4) {
            v2f av0 = *(const v2f*)(a0 + k);
            v2f av1 = *(const v2f*)(a1 + k);
            v2f bv0 = *(const v2f*)(b0 + k);
            v2f bv1 = *(const v2f*)(b1 + k);
            c00 = __builtin_amdgcn_wmma_f32_16x16x4_f32(false, av0, false, bv0, (short)0, c00, false, false);
            c01 = __builtin_amdgcn_wmma_f32_16x16x4_f32(false, av0, false, bv1, (short)0, c01, false, false);
            c10 = __builtin_amdgcn_wmma_f32_16x16x4_f32(false, av1, false, bv0, (short)0, c10, false, false);
            c11 = __builtin_amdgcn_wmma_f32_16x16x4_f32(false, av1, false, bv1, (short)0, c11, false, false);
        }
        // C/D layout: VGPR r -> row r (lanes 0-15) / r+8 (lanes 16-31); col = tile base + l16
        #pragma unroll
        for (int r = 0; r < 8; ++r) {
            #pragma unroll
            for (int s = 0; s < 2; ++s) {
                const float v0 = (s == 0 ? c00[r] : c10[r]) * TEMP_INV;  // col col0+l16
                const float v1 = (s == 0 ? c01[r] : c11[r]) * TEMP_INV;  // col col0+16+l16
                // branch-free target-logit capture (exactly one hit per row per sweep)
                tgt[s][r] += (myTgt[s][r] == col0 + l16)      ? v0 : 0.0f;
                tgt[s][r] += (myTgt[s][r] == col0 + 16 + l16) ? v1 : 0.0f;
                // online softmax over these 32 columns
                float m = fmaxf(v0, v1);
                #pragma unroll
                for (int off = 1; off < 16; off <<= 1) m = fmaxf(m, __shfl_xor(m, off, 32));
                const float nm = fmaxf(run_max[s][r], m);
                float e = __expf(v0 - nm) + __expf(v1 - nm);
                #pragma unroll
                for (int off = 1; off < 16; off <<= 1) e += __shfl_xor(e, off, 32);
                run_sum[s][r] = run_sum[s][r] * __expf(run_max[s][r] - nm) + e;
                run_max[s][r] = nm;
            }
        }
    }

    // fold target logits across each 16-lane row group, then publish per-wave stats
    #pragma unroll
    for (int s = 0; s < 2; ++s)
        #pragma unroll
        for (int r = 0; r < 8; ++r) {
            #pragma unroll
            for (int off = 1; off < 16; off <<= 1) tgt[s][r] += __shfl_xor(tgt[s][r], off, 32);
        }
    if (l16 == 0) {
        #pragma unroll
        for (int s = 0; s < 2; ++s)
            #pragma unroll
            for (int r = 0; r < 8; ++r) {
                const int m = s * 16 + r + 8 * half;
                wmax_s[wav][m] = run_max[s][r];
                wsum_s[wav][m] = run_sum[s][r];
                wtgt_s[wav][m] = tgt[s][r];
            }
    }
    __syncthreads();

    // wave 0 merges the 4 column-range partials: one row per lane (32 rows)
    if (wav == 0) {
        const int m = lane;
        float gm = wmax_s[0][m];
        #pragma unroll
        for (int w = 1; w < 4; ++w) gm = fmaxf(gm, wmax_s[w][m]);
        float gs = 0.0f, gt = 0.0f;
        #pragma unroll
        for (int w = 0; w < 4; ++w) {
            gs += wsum_s[w][m] * __expf(wmax_s[w][m] - gm);
            gt += wtgt_s[w][m];
        }
        float part = gm + __logf(gs) - gt;       // loss_i = logsumexp - target logit
        #pragma unroll
        for (int off = 1; off < 32; off <<= 1) part += __shfl_xor(part, off, 32);
        if (lane == 0) atomicAdd(loss_out, part * (1.0f / (float)NROWS));
    }
}

// ---------------------------------------------------------------------------
// Kernel 2: serial momentum scatter-update with per-row L2 renorm.
// One block per bank row; matches applied in increasing batch-index order,
// which is equivalent to the reference's serial scan (rows are independent).
// ---------------------------------------------------------------------------
__global__ __launch_bounds__(256, 1)
void mem_update_kernel(const float* __restrict__ feats,
                       const int*   __restrict__ pids,
                       const float* __restrict__ mem,
                       float*       __restrict__ out) {
    __shared__ int   list[NROWS];   // ordered matching batch indices (worst case 16 KB)
    __shared__ int   cnt_s;
    __shared__ float red[256];
    __shared__ float norm_s;

    const int y   = blockIdx.x;
    const int tid = threadIdx.x;

    if (tid < 32) {                 // wave 0 builds the ordered match list
        int count = 0;
        for (int base = 0; base < NROWS; base += 32) {
            const int  p   = pids[base + tid];
            const bool hit = (p == y);
            const unsigned int m32 = (unsigned int)__ballot(hit);   // wave32
            if (hit) {
                const int pos = count + __popc(m32 & ((1u << tid) - 1u));
                list[pos] = base + tid;
            }
            count += __popc(m32);
        }
        if (tid == 0) cnt_s = count;
    }
    __syncthreads();
    const int cnt = cnt_s;

    float v[8];
    #pragma unroll
    for (int j = 0; j < 8; ++j) v[j] = mem[(size_t)y * DDIM + tid + j * 256];

    for (int u = 0; u < cnt; ++u) {
        const float* f = feats + (size_t)list[u] * DDIM;
        float ss = 0.0f;
        #pragma unroll
        for (int j = 0; j < 8; ++j) {
            v[j] = MOM * v[j] + (1.0f - MOM) * f[tid + j * 256];
            ss += v[j] * v[j];
        }
        red[tid] = ss;
        __syncthreads();
        for (int s = 128; s > 0; s >>= 1) {
            if (tid < s) red[tid] += red[tid + s];
            __syncthreads();
        }
        if (tid == 0) norm_s = sqrtf(red[0]);
        __syncthreads();
        const float rn = 1.0f / norm_s;
        #pragma unroll
        for (int j = 0; j < 8; ++j) v[j] *= rn;
        __syncthreads();
    }

    #pragma unroll
    for (int j = 0; j < 8; ++j) out[(size_t)y * DDIM + tid + j * 256] = v[j];
}

// ---------------------------------------------------------------------------
extern "C" void kernel_launch(void* const* d_in, const int* in_sizes, int n_in,
                              void* d_out, int out_size, void* d_ws, size_t ws_size,
                              hipStream_t stream) {
    const float* feats = (const float*)d_in[0];   // [N, D] f32
    const int*   pids  = (const int*)  d_in[1];   // [N]    i32
    const float* mem   = (const float*)d_in[2];   // [L, D] f32 (unit rows)
    // d_in[3] = labels = arange(L): unused (targets == pids, segment-mean is identity)

    float* loss    = (float*)d_out;               // d_out[0]
    float* new_mem = (float*)d_out + 1;           // d_out[1 .. 1+L*D)

    zero_loss_kernel<<<1, 32, 0, stream>>>(loss);

    const size_t dynLds = (size_t)32 * ASTRIDE * sizeof(float);   // ~256.5 KB < 320 KB
    sim_loss_kernel<<<NROWS / 32, 128, dynLds, stream>>>(feats, pids, mem, loss);

    mem_update_kernel<<<LROWS, 256, 0, stream>>>(feats, pids, mem, new_mem);
}